// ECC_72593537237028
// MI455X (gfx1250) — compile-verified
//
#include <hip/hip_runtime.h>
#include <math.h>

typedef __attribute__((ext_vector_type(2))) float v2f;
typedef __attribute__((ext_vector_type(8))) float v8f;

#define NUMC   6      // classes
#define KP     48     // total prototypes
#define CDIM   512    // channels
#define HW     65536  // H*W
#define PIXB   256    // pixels per block (8 waves x 32 pixels)

static __device__ __forceinline__ v8f wmma_f32(v2f a, v2f b, v8f c) {
    // D = A(16x4,f32) * B(4x16,f32) + C(16x16,f32)
    return __builtin_amdgcn_wmma_f32_16x16x4_f32(
        /*neg_a=*/false, a, /*neg_b=*/false, b,
        /*c_mod=*/(short)0, c, /*reuse_a=*/false, /*reuse_b=*/false);
}

__global__ __launch_bounds__(256, 2)
void ecc_proto_dist_kernel(const float* __restrict__ x,
                           const float* __restrict__ proto,
                           float* __restrict__ out)
{
    // All prototypes resident for the whole kernel, interleaved as [c/2][n][2]
    // so a B-fetch (channel pair, n = lane) is one conflict-free ds_load_b64.
    __shared__ __align__(16) float lds_p[(CDIM / 2) * (KP * 2)]; // 98,304 B
    __shared__ float lds_psq[KP];
    __shared__ float lds_xsq[PIXB];

    const int tid  = threadIdx.x;
    const int lane = tid & 31;
    const int wave = tid >> 5;
    const int m    = lane & 15;   // M (pixel) / N (proto) index within 16
    const int kh   = lane >> 4;   // lane half selects K pair

    const int blockPix = blockIdx.x * PIXB;
    const int b   = blockPix >> 16;          // batch
    const int off = blockPix & (HW - 1);     // pixel offset within batch
    const int wbase = wave * 32;             // this wave's 32-pixel slice

    // ---- stage ALL prototypes into LDS (once), transposed+interleaved ----
    for (int i = tid; i < KP * CDIM; i += 256) {
        int p = i >> 9;                      // proto 0..47
        int c = i & (CDIM - 1);              // channel 0..511
        lds_p[(c >> 1) * (KP * 2) + p * 2 + (c & 1)] = proto[i];
    }
    // ---- prototype squared norms ----
    if (tid < KP) {
        const float* pr = proto + (size_t)tid * CDIM;
        float s = 0.f;
        #pragma unroll 8
        for (int c = 0; c < CDIM; ++c) { float v = pr[c]; s += v * v; }
        lds_psq[tid] = s;
    }
    __syncthreads();                         // the only block barrier

    v8f acc[2][3];
    #pragma unroll
    for (int mt = 0; mt < 2; ++mt)
        #pragma unroll
        for (int t = 0; t < 3; ++t) {
            v8f z = {};
            acc[mt][t] = z;
        }

    // Per-lane base: pixel (wbase+m) of this block, channel offset 2*kh.
    // A-operand lane layout (16x4 f32): lanes 0-15 K={0,1}, lanes 16-31 K={2,3}.
    const float* xA = x + (size_t)b * CDIM * HW + off + wbase + m
                        + (size_t)(2 * kh) * HW;

    float xsq0 = 0.f, xsq1 = 0.f;

    #pragma unroll 8
    for (int c = 0; c < CDIM; c += 4) {
        const float* pc = xA + (size_t)c * HW;
        // Four global_load_b32: channels c..c+3 for 32 pixels = four full
        // 128B cachelines, each consumed exactly once, right here.
        v2f a0, a1;
        a0.x = pc[0];        // K = c + 2*kh,     pixels wbase..wbase+15
        a0.y = pc[HW];       // K = c + 2*kh + 1
        a1.x = pc[16];       // same, pixels wbase+16..wbase+31
        a1.y = pc[HW + 16];

        xsq0 += a0.x * a0.x + a0.y * a0.y;
        xsq1 += a1.x * a1.x + a1.y * a1.y;

        // B-operand: channel pair index c/2 + kh, n = t*16 + m
        const float* pb = &lds_p[((c >> 1) + kh) * (KP * 2)];
        v2f b0 = *(const v2f*)&pb[(0 + m) * 2];
        v2f b1 = *(const v2f*)&pb[(16 + m) * 2];
        v2f b2 = *(const v2f*)&pb[(32 + m) * 2];

        acc[0][0] = wmma_f32(a0, b0, acc[0][0]);
        acc[0][1] = wmma_f32(a0, b1, acc[0][1]);
        acc[0][2] = wmma_f32(a0, b2, acc[0][2]);
        acc[1][0] = wmma_f32(a1, b0, acc[1][0]);
        acc[1][1] = wmma_f32(a1, b1, acc[1][1]);
        acc[1][2] = wmma_f32(a1, b2, acc[1][2]);
    }

    // ---- finish ||x||^2 : combine the two lane-half partial sums ----
    float s0 = xsq0 + __shfl_xor(xsq0, 16, 32);
    float s1 = xsq1 + __shfl_xor(xsq1, 16, 32);
    if (lane < 16) lds_xsq[wbase + lane] = s0;       // pixels [wbase, wbase+16)
    else           lds_xsq[wbase + 16 + m] = s1;     // pixels [wbase+16, wbase+32)
    // wave-local LDS region: no block barrier required

    // ---- epilogue: d = sqrt(max(0, xsq + psq - 2*dot)); max over 8 protos/class ----
    #pragma unroll
    for (int mt = 0; mt < 2; ++mt) {
        #pragma unroll
        for (int t = 0; t < 3; ++t) {
            float psq = lds_psq[t * 16 + m];         // proto n = t*16 + m
            v8f av = acc[mt][t];
            #pragma unroll
            for (int r = 0; r < 8; ++r) {
                int pixl = wbase + mt * 16 + r + 8 * kh;   // block-local pixel
                float xsq = lds_xsq[pixl];
                float d2 = xsq + psq - 2.0f * av[r];
                float d  = sqrtf(fmaxf(d2, 0.0f));
                // max over the 8 prototypes of this class (8-lane group)
                d = fmaxf(d, __shfl_xor(d, 1, 32));
                d = fmaxf(d, __shfl_xor(d, 2, 32));
                d = fmaxf(d, __shfl_xor(d, 4, 32));
                if ((lane & 7) == 0) {
                    int k = t * 2 + ((lane >> 3) & 1);     // class id
                    out[((size_t)b * NUMC + k) * HW + off + pixl] = d;
                }
            }
        }
    }
}

extern "C" void kernel_launch(void* const* d_in, const int* in_sizes, int n_in,
                              void* d_out, int out_size, void* d_ws, size_t ws_size,
                              hipStream_t stream) {
    const float* x     = (const float*)d_in[0];   // (4, 512, 256, 256) f32
    // d_in[1] = gt (int64) — unused in eval-mode forward
    const float* proto = (const float*)d_in[2];   // (6, 8, 512) f32
    float* out = (float*)d_out;                   // (4, 6, 256, 256) f32

    const int nPix = 4 * HW;                      // 262,144 pixels
    dim3 grid(nPix / PIXB);                       // 1024 blocks
    dim3 block(256);                              // 8 waves
    ecc_proto_dist_kernel<<<grid, block, 0, stream>>>(x, proto, out);
}